// CodeBook_38998303048173
// MI455X (gfx1250) — compile-verified
//
#include <hip/hip_runtime.h>
#include <hip/hip_bf16.h>

typedef __attribute__((ext_vector_type(2))) float v2f;
typedef __attribute__((ext_vector_type(8))) float v8f;

#define NROWS 20480   // B*W
#define DDIM  512
#define KCODES 8192
#define PDIM  4096
#define BATCH 2048
#define WNUM  10
#define TOPK  10

__device__ __forceinline__ v8f wmma4(v2f a, v2f b, v8f c) {
  // D = A(16x4 f32) * B(4x16 f32) + C(16x16 f32)
  return __builtin_amdgcn_wmma_f32_16x16x4_f32(false, a, false, b, (short)0, c,
                                               false, false);
}

__device__ __forceinline__ void topk_insert(float (&bv)[TOPK], int (&bi)[TOPK],
                                            float c, int ci) {
  if (c < bv[TOPK - 1]) {
    bv[TOPK - 1] = c;
    bi[TOPK - 1] = ci;
#pragma unroll
    for (int q = TOPK - 1; q > 0; --q) {
      if (bv[q] < bv[q - 1]) {
        float tv = bv[q]; bv[q] = bv[q - 1]; bv[q - 1] = tv;
        int   ti = bi[q]; bi[q] = bi[q - 1]; bi[q - 1] = ti;
      }
    }
  }
}

// ---------------------------------------------------------------------------
// Kernel 1: cb[K, D] = pca[K, P] @ w[D, P]^T + bias[D]
// NT GEMM, one wave computes a 64x64 tile with 16 f32 WMMA accumulators.
// ---------------------------------------------------------------------------
__global__ __launch_bounds__(128) void k_cb_gemm(const float* __restrict__ pca,
                                                 const float* __restrict__ w,
                                                 const float* __restrict__ bias,
                                                 float* __restrict__ cb) {
  const int lane = threadIdx.x & 31;
  const int wv   = threadIdx.x >> 5;
  const int gw   = blockIdx.x * 4 + wv;          // 1024 waves total
  const int tm   = (gw & 127) * 64;              // 128 tiles along K
  const int tn   = (gw >> 7) * 64;               // 8 tiles along D
  const int r    = lane & 15;
  const int hi   = lane >> 4;

  v8f acc[4][4];
#pragma unroll
  for (int i = 0; i < 4; ++i)
#pragma unroll
    for (int j = 0; j < 4; ++j) acc[i][j] = {};

  const float* aptr[4];
  const float* bptr[4];
#pragma unroll
  for (int i = 0; i < 4; ++i)
    aptr[i] = pca + (size_t)(tm + 16 * i + r) * PDIM + 2 * hi;
#pragma unroll
  for (int j = 0; j < 4; ++j)
    bptr[j] = w + (size_t)(tn + 16 * j + r) * PDIM + 2 * hi;

  for (int p = 0; p < PDIM; p += 4) {
    v2f a[4], bf[4];
#pragma unroll
    for (int i = 0; i < 4; ++i) a[i] = *(const v2f*)(aptr[i] + p);
#pragma unroll
    for (int j = 0; j < 4; ++j) bf[j] = *(const v2f*)(bptr[j] + p);
#pragma unroll
    for (int i = 0; i < 4; ++i)
#pragma unroll
      for (int j = 0; j < 4; ++j) acc[i][j] = wmma4(a[i], bf[j], acc[i][j]);
  }

#pragma unroll
  for (int j = 0; j < 4; ++j) {
    const float bj = bias[tn + 16 * j + r];
#pragma unroll
    for (int i = 0; i < 4; ++i) {
#pragma unroll
      for (int v = 0; v < 8; ++v) {
        const int m = tm + 16 * i + v + 8 * hi;   // C layout: VGPR v -> M=v (+8 for hi half)
        cb[(size_t)m * DDIM + tn + 16 * j + r] = acc[i][j][v] + bj;
      }
    }
  }
}

// ---------------------------------------------------------------------------
// Kernel 2: cbn[k] = sum_d cb[k,d]^2   (one wave per row)
// ---------------------------------------------------------------------------
__global__ __launch_bounds__(256) void k_cbnorm(const float* __restrict__ cb,
                                                float* __restrict__ cbn) {
  const int lane = threadIdx.x & 31;
  const int wv   = threadIdx.x >> 5;
  const int row  = blockIdx.x * 8 + wv;
  const float* p = cb + (size_t)row * DDIM;
  float s = 0.f;
#pragma unroll
  for (int i = 0; i < DDIM / 32; ++i) {
    float v = p[lane + i * 32];
    s += v * v;
  }
#pragma unroll
  for (int off = 16; off > 0; off >>= 1) s += __shfl_xor(s, off, 32);
  if (lane == 0) cbn[row] = s;
}

// ---------------------------------------------------------------------------
// Kernel 3: fused score GEMM + per-row top-10.
// score[n,k] = cbn[k] - 2 * dot(z_e[n], cb[k])  (row-constant |z|^2 dropped)
// Block = 4 waves; each wave owns 32 rows x 64 cols per K-chunk
// (8 WMMA accumulators; 6 loads per 8 WMMAs in the d-loop).
// Score tile staged in LDS (row stride 65 -> conflict-free scans);
// each lane owns exactly one row, so no cross-lane merge is needed.
// ---------------------------------------------------------------------------
__global__ __launch_bounds__(128) void k_score_topk(const float* __restrict__ ze,
                                                    const float* __restrict__ cb,
                                                    const float* __restrict__ cbn,
                                                    int* __restrict__ tidx) {
  __shared__ float lds[4][32 * 65];
  const int lane    = threadIdx.x & 31;
  const int wv      = threadIdx.x >> 5;
  const int r       = lane & 15;
  const int hi      = lane >> 4;
  const int rowBase = blockIdx.x * 128 + wv * 32;

  float bv[TOPK];
  int   bi[TOPK];
#pragma unroll
  for (int j = 0; j < TOPK; ++j) { bv[j] = __builtin_inff(); bi[j] = 0; }

  const float* aPtr[2];
#pragma unroll
  for (int i = 0; i < 2; ++i)
    aPtr[i] = ze + (size_t)(rowBase + 16 * i + r) * DDIM + 2 * hi;

  for (int kc = 0; kc < KCODES; kc += 64) {
    if (kc + 64 < KCODES) __builtin_prefetch(cb + (size_t)(kc + 64) * DDIM, 0, 1);

    v8f acc[2][4];
#pragma unroll
    for (int i = 0; i < 2; ++i)
#pragma unroll
      for (int j = 0; j < 4; ++j) acc[i][j] = {};

    const float* bPtr[4];
#pragma unroll
    for (int j = 0; j < 4; ++j)
      bPtr[j] = cb + (size_t)(kc + 16 * j + r) * DDIM + 2 * hi;

    for (int d = 0; d < DDIM; d += 4) {
      v2f a[2];
#pragma unroll
      for (int i = 0; i < 2; ++i) a[i] = *(const v2f*)(aPtr[i] + d);
#pragma unroll
      for (int j = 0; j < 4; ++j) {
        v2f b = *(const v2f*)(bPtr[j] + d);
#pragma unroll
        for (int i = 0; i < 2; ++i) acc[i][j] = wmma4(a[i], b, acc[i][j]);
      }
    }

    // score -> LDS tile [32 rows][64 cols], row stride 65
#pragma unroll
    for (int j = 0; j < 4; ++j) {
      const float cn = cbn[kc + 16 * j + r];
#pragma unroll
      for (int i = 0; i < 2; ++i) {
#pragma unroll
        for (int v = 0; v < 8; ++v) {
          const int m = 16 * i + v + 8 * hi;
          lds[wv][m * 65 + 16 * j + r] = cn - 2.0f * acc[i][j][v];
        }
      }
    }
    __syncthreads();

    // lane l scans row l (bank = (l + t) % 64 -> conflict-free)
    const float* rowp = &lds[wv][lane * 65];
#pragma unroll 4
    for (int t = 0; t < 64; ++t) {
      topk_insert(bv, bi, rowp[t], kc + t);
    }
    __syncthreads();
  }

  const int n = rowBase + lane;
#pragma unroll
  for (int j = 0; j < TOPK; ++j) tidx[n * TOPK + j] = bi[j];
}

// ---------------------------------------------------------------------------
// Kernel 4: greedy selection with exclusion, one thread per batch row.
// Candidate lists are sorted ascending, so first unused entry == masked argmin.
// ---------------------------------------------------------------------------
__global__ void k_greedy(const int* __restrict__ tidx, int* __restrict__ idxf) {
  const int b = blockIdx.x * blockDim.x + threadIdx.x;
  if (b >= BATCH) return;
  int used[WNUM];
  for (int w = 0; w < WNUM; ++w) {
    const int base = (b * WNUM + w) * TOPK;
    int pick = tidx[base];
    bool done = false;
    for (int j = 0; j < TOPK && !done; ++j) {
      const int cand = tidx[base + j];
      bool ok = true;
      for (int u = 0; u < w; ++u) ok = ok && (used[u] != cand);
      if (ok) { pick = cand; done = true; }
    }
    used[w] = pick;
    idxf[b * WNUM + w] = pick;
  }
}

// ---------------------------------------------------------------------------
// Kernel 5: gather z_q (forward value of straight-through output) and
// per-block partial of sum((z_q - z_e)^2). One block per row.
// ---------------------------------------------------------------------------
__global__ __launch_bounds__(128) void k_gather(const float* __restrict__ ze,
                                                const float* __restrict__ cb,
                                                const int* __restrict__ idxf,
                                                float* __restrict__ out,
                                                float* __restrict__ partial) {
  __shared__ float wsum[4];
  const int n = blockIdx.x;
  const int t = threadIdx.x;
  const int k = idxf[n];
  const float* crow = cb + (size_t)k * DDIM;
  const float* zrow = ze + (size_t)n * DDIM;
  float s = 0.f;
#pragma unroll
  for (int i = 0; i < DDIM / 128; ++i) {
    const int d = t + i * 128;
    const float q = crow[d];
    const float diff = q - zrow[d];
    s += diff * diff;
    out[(size_t)n * DDIM + d] = q;
  }
#pragma unroll
  for (int off = 16; off > 0; off >>= 1) s += __shfl_xor(s, off, 32);
  if ((t & 31) == 0) wsum[t >> 5] = s;
  __syncthreads();
  if (t == 0) partial[n] = wsum[0] + wsum[1] + wsum[2] + wsum[3];
}

// Kernel 6: deterministic final loss reduction.
__global__ __launch_bounds__(256) void k_loss_final(const float* __restrict__ partial,
                                                    float* __restrict__ lossSlot) {
  __shared__ float red[256];
  float s = 0.f;
  for (int i = threadIdx.x; i < NROWS; i += 256) s += partial[i];
  red[threadIdx.x] = s;
  __syncthreads();
  for (int off = 128; off > 0; off >>= 1) {
    if ((int)threadIdx.x < off) red[threadIdx.x] += red[threadIdx.x + off];
    __syncthreads();
  }
  if (threadIdx.x == 0)
    lossSlot[0] = red[0] * (1.0f / ((float)NROWS * (float)DDIM));
}

// ---------------------------------------------------------------------------
extern "C" void kernel_launch(void* const* d_in, const int* in_sizes, int n_in,
                              void* d_out, int out_size, void* d_ws,
                              size_t ws_size, hipStream_t stream) {
  const float* ze   = (const float*)d_in[0];  // [20480, 512]
  const float* pca  = (const float*)d_in[1];  // [8192, 4096]
  const float* w    = (const float*)d_in[2];  // [512, 4096]
  const float* bias = (const float*)d_in[3];  // [512]

  float* out      = (float*)d_out;            // [20480*512] z_q  + [1] loss
  float* lossSlot = out + (size_t)NROWS * DDIM;

  // workspace layout (32-bit words)
  float* ws_f    = (float*)d_ws;
  float* cb      = ws_f;                                  // 8192*512
  float* cbn     = ws_f + (size_t)KCODES * DDIM;          // 8192
  int*   tidx    = (int*)(cbn + KCODES);                  // 20480*10
  int*   idxf    = tidx + (size_t)NROWS * TOPK;           // 20480
  float* partial = (float*)(idxf + NROWS);                // 20480

  // 1) cb = pca @ w^T + bias : 1024 waves of 64x64 tiles
  k_cb_gemm<<<256, 128, 0, stream>>>(pca, w, bias, cb);
  // 2) row norms of cb
  k_cbnorm<<<KCODES / 8, 256, 0, stream>>>(cb, cbn);
  // 3) fused score GEMM + top-10 (128 rows per block, 32 rows per wave)
  k_score_topk<<<NROWS / 128, 128, 0, stream>>>(ze, cb, cbn, tidx);
  // 4) greedy masked argmin
  k_greedy<<<(BATCH + 255) / 256, 256, 0, stream>>>(tidx, idxf);
  // 5) gather + partial loss
  k_gather<<<NROWS, 128, 0, stream>>>(ze, cb, idxf, out, partial);
  // 6) deterministic loss
  k_loss_final<<<1, 256, 0, stream>>>(partial, lossSlot);
}